// DAGERC_44727789421059
// MI455X (gfx1250) — compile-verified
//
#include <hip/hip_runtime.h>

#define Bn   32
#define Nn   128
#define EMBn 1024
#define HIDn 512
#define Ln   4
#define NCn  7
#define HCAT (HIDn * (Ln + 1) + EMBn)   // 3584
#define G3   (3 * HIDn)                 // 1536
#define ROWS ((size_t)Bn * Nn)          // 4096
#define SCAN_BLOCKS 48
#define SCAN_THREADS 128

// LDS layout for scan kernel (floats). Row stride 516 => bank = (4*r16+koff)%64,
// conflict-free for the 16-lane half-wave b64 access pattern.
#define WSTRIDE 516
#define WHH_FLOATS (4 * 16 * WSTRIDE)            // 33024 floats = 132,096 B
#define MOFF       WHH_FLOATS
#define SMEM_FLOATS (WHH_FLOATS + 16 * WSTRIDE)  // + 33,024 B -> 165,120 B total

typedef float v2f __attribute__((ext_vector_type(2)));
typedef float v8f __attribute__((ext_vector_type(8)));

__device__ __forceinline__ v8f wmma_f32(v2f a, v2f b, v8f c) {
  // V_WMMA_F32_16X16X4_F32: full-precision fp32 matrix op (CDNA5)
  return __builtin_amdgcn_wmma_f32_16x16x4_f32(
      false, a, false, b, (short)0, c, false, false);
}

__device__ __forceinline__ float sigmoidf_(float x) {
  return 1.0f / (1.0f + expf(-x));
}

// ---------------------------------------------------------------------------
// Parallel WMMA GEMM, 16x64 strip per wave (1 A-frag feeds 4 WMMAs per k-step):
// C[m,n] = act( sum_k A[m*lda+k] * W[n*ldw+k] + bias[n] )
// ---------------------------------------------------------------------------
template <int RELU>
__global__ void __launch_bounds__(128) gemm_wmma_kernel(
    const float* __restrict__ A, int lda,
    const float* __restrict__ W, int ldw,
    const float* __restrict__ bias,
    float* __restrict__ C, int ldc,
    int Mt, int Nstrips, int K) {
  int wave = blockIdx.x * 4 + (threadIdx.x >> 5);
  if (wave >= Mt * Nstrips) return;
  int tm = (wave / Nstrips) << 4;
  int tn = (wave % Nstrips) << 6;       // 64-wide strip = 4 tiles
  int lane = threadIdx.x & 31;
  int r16  = lane & 15;
  int koff = (lane >> 4) << 1;          // lanes 0-15 -> K{0,1}; 16-31 -> K{2,3}
  const float* ap  = A + (size_t)(tm + r16) * lda + koff;
  const float* bp0 = W + (size_t)(tn + r16) * ldw + koff;
  const float* bp1 = bp0 + (size_t)16 * ldw;
  const float* bp2 = bp0 + (size_t)32 * ldw;
  const float* bp3 = bp0 + (size_t)48 * ldw;
  v8f acc0 = {}, acc1 = {}, acc2 = {}, acc3 = {};
#pragma unroll 2
  for (int k = 0; k < K; k += 4) {
    v2f av = *(const v2f*)(ap + k);
    v2f b0 = *(const v2f*)(bp0 + k);
    v2f b1 = *(const v2f*)(bp1 + k);
    v2f b2 = *(const v2f*)(bp2 + k);
    v2f b3 = *(const v2f*)(bp3 + k);
    acc0 = wmma_f32(av, b0, acc0);
    acc1 = wmma_f32(av, b1, acc1);
    acc2 = wmma_f32(av, b2, acc2);
    acc3 = wmma_f32(av, b3, acc3);
  }
  int rbase = tm + ((lane >> 4) << 3);  // C/D layout: lanes>=16 hold rows +8
  v8f accs[4] = {acc0, acc1, acc2, acc3};
#pragma unroll
  for (int n = 0; n < 4; ++n) {
    int col = tn + (n << 4) + r16;
    float bb = bias ? bias[col] : 0.0f;
#pragma unroll
    for (int r = 0; r < 8; ++r) {
      float v = accs[n][r] + bb;
      if (RELU) v = fmaxf(v, 0.0f);
      C[(size_t)(rbase + r) * ldc + col] = v;
    }
  }
}

// ---------------------------------------------------------------------------
// Utility kernels
// ---------------------------------------------------------------------------
__global__ void init_counters_kernel(unsigned* c) {
  if (threadIdx.x < 8) c[threadIdx.x] = 0u;
}

__global__ void copy_features_kernel(const float4* __restrict__ f, float* __restrict__ Hcat) {
  size_t idx = (size_t)blockIdx.x * blockDim.x + threadIdx.x;  // over ROWS*EMB/4
  if (idx >= ROWS * (EMBn / 4)) return;
  size_t row = idx / (EMBn / 4);
  size_t c4  = idx % (EMBn / 4);
  ((float4*)(Hcat + row * HCAT + (size_t)5 * HIDn))[c4] = f[idx];
}

__global__ void mlp2_kernel(const float* __restrict__ A, const float* __restrict__ W,
                            const float* __restrict__ bias, float* __restrict__ out) {
  int o = blockIdx.x * blockDim.x + threadIdx.x;
  if (o >= (int)(ROWS * NCn)) return;
  int row = o / NCn, c = o % NCn;
  float s = bias[c];
  const float* a = A + (size_t)row * HIDn;
  const float* w = W + (size_t)c * HIDn;
  for (int k = 0; k < HIDn; ++k) s += a[k] * w[k];
  out[o] = s;
}

// ---------------------------------------------------------------------------
// Grid barrier: monotonic counter (never reset -> no sense-reversal race).
// ---------------------------------------------------------------------------
__device__ __forceinline__ void grid_barrier(unsigned* cnt, unsigned target) {
  __syncthreads();
  if (threadIdx.x == 0) {
    __threadfence();
    atomicAdd(cnt, 1u);
    while (atomicAdd(cnt, 0u) < target) __builtin_amdgcn_s_sleep(2);
    __threadfence();
  }
  __syncthreads();
}

// wave32 shuffle reductions + tiny LDS combine (2 __syncthreads total)
__device__ __forceinline__ float wave_sum(float v) {
#pragma unroll
  for (int off = 16; off; off >>= 1) v += __shfl_xor(v, off, 32);
  return v;
}
__device__ __forceinline__ float wave_max(float v) {
#pragma unroll
  for (int off = 16; off; off >>= 1) v = fmaxf(v, __shfl_xor(v, off, 32));
  return v;
}
__device__ __forceinline__ float block_reduce_sum(float v, float* s4) {
  v = wave_sum(v);
  if ((threadIdx.x & 31) == 0) s4[threadIdx.x >> 5] = v;
  __syncthreads();
  float r = s4[0] + s4[1] + s4[2] + s4[3];
  __syncthreads();
  return r;
}
__device__ __forceinline__ float block_reduce_max(float v, float* s4) {
  v = wave_max(v);
  if ((threadIdx.x & 31) == 0) s4[threadIdx.x >> 5] = v;
  __syncthreads();
  float r = fmaxf(fmaxf(s4[0], s4[1]), fmaxf(s4[2], s4[3]));
  __syncthreads();
  return r;
}

// ---------------------------------------------------------------------------
// Persistent scan kernel: one launch per layer. 48 blocks x 128 threads.
// Blocks 0..31: per-batch softmax/M/GRU (batch = blockIdx.x).
// All 48 blocks (192 waves): WMMA GEMM gh = M(32x512) @ whh^T(1536x512),
// with whh tiles staged ONCE into LDS (320KB/WGP on CDNA5) and M staged
// per step -> the 128-iteration WMMA loop runs entirely out of LDS.
// ---------------------------------------------------------------------------
__global__ void __launch_bounds__(SCAN_THREADS) scan_layer_kernel(
    const float* __restrict__ Hprev,  // Hcat + l*HID   (row stride HCAT)
    float* __restrict__ Hout,         // Hcat + (l+1)*HID
    const float* __restrict__ gi,     // [B,N,1536]  (= Hprev@wih^T + bih)
    const float* __restrict__ adj,    // [B,N,N]
    const float* __restrict__ gwq,    // gat_w[l][0:512]
    const float* __restrict__ gwk,    // gat_w[l][512:1024]
    const float* __restrict__ gbp,    // &gat_b[l]
    const float* __restrict__ whh,    // [1536,512]
    const float* __restrict__ bhh,    // [1536]
    float* __restrict__ Mbuf,         // [B,512]   workspace
    float* __restrict__ ghbuf,        // [B,1536]  workspace
    unsigned* __restrict__ cnt) {
  extern __shared__ float smem[];     // [WHH_FLOATS] whh tiles | [16*WSTRIDE] M tile
  __shared__ float s_qproj[Nn];
  __shared__ float s_kproj[Nn];
  __shared__ float s_w[SCAN_THREADS];
  __shared__ float s_red[4];

  const int b = blockIdx.x;
  const int t = threadIdx.x;
  const float gb = gbp[0];
  unsigned epoch = 0;

  // ---- stage this block's 4 whh tiles (64 rows x 512) into LDS, once
  {
    for (int idx = t; idx < 4 * 16 * (HIDn / 4); idx += SCAN_THREADS) {
      int w   = idx / (16 * (HIDn / 4));
      int rem = idx % (16 * (HIDn / 4));
      int row = rem >> 7;               // HIDn/4 = 128 float4 per row
      int c4  = rem & 127;
      int tnw = ((b * 4 + w) % (G3 / 16)) << 4;
      float4 v = ((const float4*)(whh + (size_t)(tnw + row) * HIDn))[c4];
      ((float4*)(smem + (size_t)(w * 16 + row) * WSTRIDE))[c4] = v;
    }
    __syncthreads();
  }

  // ---- init: qproj for all rows, h0 = GRU(Hprev[:,0], h=0), kproj[0]
  if (b < Bn) {
    const float* Hp = Hprev + (size_t)b * Nn * HCAT;
    float* Ho = Hout + (size_t)b * Nn * HCAT;
    for (int i = t; i < Nn; i += SCAN_THREADS) {
      const float* row = Hp + (size_t)i * HCAT;
      float s = 0.f;
      for (int k = 0; k < HIDn; ++k) s += row[k] * gwq[k];
      s_qproj[i] = s;
    }
    const float* gi0 = gi + (size_t)b * Nn * G3;  // row 0
    float kp = 0.f;
#pragma unroll
    for (int q = 0; q < HIDn / SCAN_THREADS; ++q) {
      int d = t + q * SCAN_THREADS;
      float r = sigmoidf_(gi0[d] + bhh[d]);
      float z = sigmoidf_(gi0[HIDn + d] + bhh[HIDn + d]);
      float n = tanhf(gi0[2 * HIDn + d] + r * bhh[2 * HIDn + d]);
      float h = (1.f - z) * n;  // h_prev = 0
      Ho[d] = h;
      kp += h * gwk[d];
    }
    float kps = block_reduce_sum(kp, s_red);
    if (t == 0) s_kproj[0] = kps;
    __syncthreads();
  }

  for (int i = 1; i < Nn; ++i) {
    if (b < Bn) {
      float* Ho = Hout + (size_t)b * Nn * HCAT;
      const float* gib = gi + (size_t)b * Nn * G3;
      // prefetch gi row i (used by next iteration's GRU) into L2/WGP$
      __builtin_prefetch(gib + (size_t)i * G3 + (size_t)t * 12, 0, 0);
      // (a) finish GRU for row i-1 (needs ghbuf from previous step's GEMM)
      if (i >= 2) {
        int ip = i - 1;
        const float* gir = gib + (size_t)ip * G3;
        float kp = 0.f;
#pragma unroll
        for (int q = 0; q < HIDn / SCAN_THREADS; ++q) {
          int d = t + q * SCAN_THREADS;
          float Mv = Mbuf[b * HIDn + d];  // M from step i-1 (h_prev for GRUCell)
          float r = sigmoidf_(gir[d] + ghbuf[b * G3 + d]);
          float z = sigmoidf_(gir[HIDn + d] + ghbuf[b * G3 + HIDn + d]);
          float n = tanhf(gir[2 * HIDn + d] + r * ghbuf[b * G3 + 2 * HIDn + d]);
          float h = (1.f - z) * n + z * Mv;
          Ho[(size_t)ip * HCAT + d] = h;
          kp += h * gwk[d];
        }
        float kps = block_reduce_sum(kp, s_red);
        if (t == 0) s_kproj[ip] = kps;
        __syncthreads();
      }
      // (b) masked softmax over j < i  (thread t = key j)
      float alpha;
      if (t < i) {
        float mask = adj[((size_t)b * Nn + i) * Nn + t];
        alpha = s_qproj[i] + s_kproj[t] + gb - (1.f - mask) * 1e30f;
      } else {
        alpha = -3.0e38f;
      }
      float amax = block_reduce_max(alpha, s_red);
      float e = (t < i) ? expf(alpha - amax) : 0.f;
      float esum = block_reduce_sum(e, s_red);
      s_w[t] = e / esum;
      __syncthreads();
      // (c) M[b,:] = sum_{j<i} w[j] * H1[b,j,:]
#pragma unroll
      for (int q = 0; q < HIDn / SCAN_THREADS; ++q) {
        int d = t + q * SCAN_THREADS;
        float acc = 0.f;
        for (int j = 0; j < i; ++j) acc += s_w[j] * Ho[(size_t)j * HCAT + d];
        Mbuf[b * HIDn + d] = acc;
      }
    }
    ++epoch; grid_barrier(cnt, epoch * SCAN_BLOCKS);

    // ---- GEMM phase: gh = M(32x512) @ whh^T + bhh -> (32x1536)
    // 192 waves, one 16x16 tile each; A and B both from LDS.
    {
      int w    = t >> 5;
      int wave = b * 4 + w;
      int tm   = (wave / (G3 / 16)) << 4;   // uniform per block (96 % 4 == 0)
      int tn   = (wave % (G3 / 16)) << 4;
      // stage M rows [tm, tm+16) into LDS (32KB)
      for (int idx = t; idx < 16 * (HIDn / 4); idx += SCAN_THREADS) {
        int row = idx >> 7, c4 = idx & 127;
        float4 v = ((const float4*)(Mbuf + (size_t)(tm + row) * HIDn))[c4];
        ((float4*)(smem + MOFF + (size_t)row * WSTRIDE))[c4] = v;
      }
      __syncthreads();
      int lane = t & 31;
      int r16  = lane & 15;
      int koff = (lane >> 4) << 1;
      const float* ap = smem + MOFF + (size_t)r16 * WSTRIDE + koff;
      const float* bp = smem + (size_t)(w * 16 + r16) * WSTRIDE + koff;
      v8f acc = {};
#pragma unroll 8
      for (int k = 0; k < HIDn; k += 4) {
        v2f av = *(const v2f*)(ap + k);
        v2f bv = *(const v2f*)(bp + k);
        acc = wmma_f32(av, bv, acc);
      }
      int col   = tn + r16;
      int rbase = tm + ((lane >> 4) << 3);
      float bb = bhh[col];
#pragma unroll
      for (int r = 0; r < 8; ++r)
        ghbuf[(size_t)(rbase + r) * G3 + col] = acc[r] + bb;
      __syncthreads();   // M-tile in LDS reused next step; keep phases clean
    }
    ++epoch; grid_barrier(cnt, epoch * SCAN_BLOCKS);
  }

  // ---- final GRU for row N-1
  if (b < Bn) {
    int ip = Nn - 1;
    float* Ho = Hout + (size_t)b * Nn * HCAT;
    const float* gir = gi + ((size_t)b * Nn + ip) * G3;
#pragma unroll
    for (int q = 0; q < HIDn / SCAN_THREADS; ++q) {
      int d = t + q * SCAN_THREADS;
      float Mv = Mbuf[b * HIDn + d];
      float r = sigmoidf_(gir[d] + ghbuf[b * G3 + d]);
      float z = sigmoidf_(gir[HIDn + d] + ghbuf[b * G3 + HIDn + d]);
      float n = tanhf(gir[2 * HIDn + d] + r * ghbuf[b * G3 + 2 * HIDn + d]);
      Ho[(size_t)ip * HCAT + d] = (1.f - z) * n + z * Mv;
    }
  }
}

// ---------------------------------------------------------------------------
extern "C" void kernel_launch(void* const* d_in, const int* in_sizes, int n_in,
                              void* d_out, int out_size, void* d_ws, size_t ws_size,
                              hipStream_t stream) {
  (void)in_sizes; (void)n_in; (void)out_size; (void)ws_size;
  const float* features = (const float*)d_in[0];
  const float* adj      = (const float*)d_in[1];
  /* d_in[2] s_mask: unused (no_rel_attn) */
  const float* fc1_w    = (const float*)d_in[3];
  const float* fc1_b    = (const float*)d_in[4];
  const float* gat_w    = (const float*)d_in[5];   // (4, 1024)
  const float* gat_b    = (const float*)d_in[6];   // (4,)
  const float* gru_wih  = (const float*)d_in[7];   // (4, 1536, 512)
  const float* gru_whh  = (const float*)d_in[8];   // (4, 1536, 512)
  const float* gru_bih  = (const float*)d_in[9];   // (4, 1536)
  const float* gru_bhh  = (const float*)d_in[10];  // (4, 1536)
  const float* mlp_w0   = (const float*)d_in[11];  // (512, 3584)
  const float* mlp_b0   = (const float*)d_in[12];
  const float* mlp_w1   = (const float*)d_in[13];  // (512, 512)
  const float* mlp_b1   = (const float*)d_in[14];
  const float* mlp_w2   = (const float*)d_in[15];  // (7, 512)
  const float* mlp_b2   = (const float*)d_in[16];
  float* out = (float*)d_out;

  // workspace layout (floats); everything rewritten every call (determinism)
  float* ws    = (float*)d_ws;
  float* Hcat  = ws;                         // ROWS * HCAT  (H0..H4 | features)
  float* gibuf = Hcat + ROWS * HCAT;         // ROWS * G3    (reused for MLP temps)
  float* Mbuf  = gibuf + ROWS * G3;          // B * HID
  float* ghbuf = Mbuf + (size_t)Bn * HIDn;   // B * G3
  unsigned* cnt = (unsigned*)(ghbuf + (size_t)Bn * G3);  // 8 barrier counters
  float* mlph0 = gibuf;                      // reuse gi region after scans
  float* mlph1 = gibuf + ROWS * HIDn;

  init_counters_kernel<<<1, 32, 0, stream>>>(cnt);

  // features -> Hcat[:, 2560:3584]
  copy_features_kernel<<<(unsigned)((ROWS * (EMBn / 4) + 255) / 256), 256, 0, stream>>>(
      (const float4*)features, Hcat);

  // H0 = relu(features @ fc1_w^T + fc1_b) -> Hcat[:, 0:512]
  {
    int Mt = (int)(ROWS / 16), Ns = HIDn / 64;
    gemm_wmma_kernel<1><<<(Mt * Ns + 3) / 4, 128, 0, stream>>>(
        features, EMBn, fc1_w, EMBn, fc1_b, Hcat, HCAT, Mt, Ns, EMBn);
  }

  const size_t SMEM_BYTES = (size_t)SMEM_FLOATS * sizeof(float);  // ~165 KB < 320 KB/WGP
  for (int l = 0; l < Ln; ++l) {
    const float* Hprev = Hcat + (size_t)l * HIDn;
    float* Hout = Hcat + (size_t)(l + 1) * HIDn;
    // gi = Hprev @ wih^T + bih  (batched across all (b,i) rows)
    {
      int Mt = (int)(ROWS / 16), Ns = G3 / 64;
      gemm_wmma_kernel<0><<<(Mt * Ns + 3) / 4, 128, 0, stream>>>(
          Hprev, HCAT, gru_wih + (size_t)l * G3 * HIDn, HIDn,
          gru_bih + (size_t)l * G3, gibuf, G3, Mt, Ns, HIDn);
    }
    // sequential DAG scan (persistent kernel, grid-wide barriers)
    scan_layer_kernel<<<SCAN_BLOCKS, SCAN_THREADS, SMEM_BYTES, stream>>>(
        Hprev, Hout, gibuf, adj,
        gat_w + (size_t)l * 2 * HIDn, gat_w + (size_t)l * 2 * HIDn + HIDn,
        gat_b + l,
        gru_whh + (size_t)l * G3 * HIDn, gru_bhh + (size_t)l * G3,
        Mbuf, ghbuf, cnt + l);
  }

  // MLP head
  {
    int Mt = (int)(ROWS / 16), Ns = HIDn / 64;
    gemm_wmma_kernel<1><<<(Mt * Ns + 3) / 4, 128, 0, stream>>>(
        Hcat, HCAT, mlp_w0, HCAT, mlp_b0, mlph0, HIDn, Mt, Ns, HCAT);
    gemm_wmma_kernel<1><<<(Mt * Ns + 3) / 4, 128, 0, stream>>>(
        mlph0, HIDn, mlp_w1, HIDn, mlp_b1, mlph1, HIDn, Mt, Ns, HIDn);
  }
  mlp2_kernel<<<(unsigned)((ROWS * NCn + 255) / 256), 256, 0, stream>>>(
      mlph1, mlp_w2, mlp_b2, out);
}